// MinMaxRelOut_25374666784831
// MI455X (gfx1250) — compile-verified
//
#include <hip/hip_runtime.h>

#define B_DIM 1024
#define D_DIM 512
#define O_DIM 256

typedef __attribute__((ext_vector_type(2))) float v2f;
typedef __attribute__((ext_vector_type(8))) float v8f;

// ---------------------------------------------------------------------------
// Kernel 0: column sums via V_WMMA_F32_16X16X4_F32 with all-ones A.
//   C[m,n] += sum_k A[m,k]*B[k,n] ; A==1 -> C[m,n] = sum_k B[k,n] (all m equal)
//   blocks 0..31  : Sx[d] = sum_b (1 - x[b,d])   (16 d-columns per wave)
//   blocks 32..47 : Sw[o] = sum_d (1 - w[d,o])   (16 o-columns per wave)
// One wave (32 threads) per block -> EXEC is all-ones as WMMA requires.
// ---------------------------------------------------------------------------
__global__ void mmro_sums_wmma(const float* __restrict__ x,
                               const float* __restrict__ w,
                               float* __restrict__ Sx,
                               float* __restrict__ Sw) {
    const int lane = threadIdx.x;        // 0..31
    const int grp  = lane >> 4;          // 0 or 1
    const int col  = lane & 15;          // N = lane % 16
    v2f a; a[0] = 1.0f; a[1] = 1.0f;     // all-ones A (16x4 f32)
    v8f c = {};
    if (blockIdx.x < 32) {
        const int dbase = blockIdx.x * 16;
        for (int b0 = 0; b0 < B_DIM; b0 += 4) {
            const int r0 = b0 + (grp << 1);
            v2f bb;
            bb[0] = 1.0f - x[(size_t)r0 * D_DIM + dbase + col];
            bb[1] = 1.0f - x[(size_t)(r0 + 1) * D_DIM + dbase + col];
            c = __builtin_amdgcn_wmma_f32_16x16x4_f32(false, a, false, bb,
                                                      (short)0, c, false, false);
        }
        if (lane < 16) Sx[dbase + lane] = c[0];
    } else {
        const int obase = (blockIdx.x - 32) * 16;
        for (int d0 = 0; d0 < D_DIM; d0 += 4) {
            const int r0 = d0 + (grp << 1);
            v2f bb;
            bb[0] = 1.0f - w[(size_t)r0 * O_DIM + obase + col];
            bb[1] = 1.0f - w[(size_t)(r0 + 1) * O_DIM + obase + col];
            c = __builtin_amdgcn_wmma_f32_16x16x4_f32(false, a, false, bb,
                                                      (short)0, c, false, false);
        }
        if (lane < 16) Sw[obase + lane] = c[0];
    }
}

// ---------------------------------------------------------------------------
// Kernel 1: rel_x[d,o] = 1 - (sum_b min(1-x[b,d], 1-t[b,o])) / Sx[d]
// blockDim = 128 (one d per thread), 8 o-columns per thread in registers.
// t-chunk (16 b x 8 o) staged in LDS per step; x reads coalesced over d-lanes.
// grid = (D/128) * (O/8) = 4 * 32 = 128 blocks.
// ---------------------------------------------------------------------------
__global__ void mmro_relx(const float* __restrict__ x,
                          const float* __restrict__ t,
                          const float* __restrict__ Sx,
                          float* __restrict__ relx) {
    __shared__ float tc[16][8];
    const int tid = threadIdx.x;                 // 0..127
    const int db  = blockIdx.x & 3;
    const int ob  = blockIdx.x >> 2;
    const int d   = db * 128 + tid;
    const int o0  = ob * 8;

    float acc[8];
#pragma unroll
    for (int j = 0; j < 8; ++j) acc[j] = 0.0f;

    for (int b0 = 0; b0 < B_DIM; b0 += 16) {
        // stage 16x8 t-complement tile: 128 threads, one element each
        const int i = tid >> 3, j = tid & 7;
        tc[i][j] = 1.0f - t[(size_t)(b0 + i) * O_DIM + o0 + j];
        __syncthreads();
#pragma unroll 4
        for (int i2 = 0; i2 < 16; ++i2) {
            const float xcv = 1.0f - x[(size_t)(b0 + i2) * D_DIM + d];
#pragma unroll
            for (int j2 = 0; j2 < 8; ++j2)
                acc[j2] += fminf(xcv, tc[i2][j2]);
        }
        __syncthreads();
    }
    const float sxd = Sx[d];
#pragma unroll
    for (int j = 0; j < 8; ++j)
        relx[(size_t)d * O_DIM + o0 + j] = 1.0f - acc[j] / sxd;
}

// ---------------------------------------------------------------------------
// Kernel 2: rel_w[b,o] = 1 - (sum_d min(1-w[d,o], 1-t[b,o])) / Sw[o]
// blockDim = 256 (o = tid, covers full O), 16 b-rows per thread in registers.
// w row reads coalesced over o-lanes; grid = B/16 = 64 blocks.
// ---------------------------------------------------------------------------
__global__ void mmro_relw(const float* __restrict__ w,
                          const float* __restrict__ t,
                          const float* __restrict__ Sw,
                          float* __restrict__ relw) {
    const int o  = threadIdx.x;      // 0..255
    const int b0 = blockIdx.x * 16;

    float tcv[16], acc[16];
#pragma unroll
    for (int i = 0; i < 16; ++i) {
        tcv[i] = 1.0f - t[(size_t)(b0 + i) * O_DIM + o];
        acc[i] = 0.0f;
    }
    for (int d = 0; d < D_DIM; ++d) {
        const float wcv = 1.0f - w[(size_t)d * O_DIM + o];
#pragma unroll
        for (int i = 0; i < 16; ++i)
            acc[i] += fminf(wcv, tcv[i]);
    }
    const float swo = Sw[o];
#pragma unroll
    for (int i = 0; i < 16; ++i)
        relw[(size_t)(b0 + i) * O_DIM + o] = 1.0f - acc[i] / swo;
}

// ---------------------------------------------------------------------------
// Kernel 3: fused argmin passes + output gather.
//   bestx = min_d max(x[b,d], rel_x[d,o])   (first index -> strict < update)
//   bestw = min_d max(rel_w[b,o], w[d,o])
//   chosen value at the running argmin: max(x[b,d], w[d,o])  (no re-gather)
// blockDim = 256 (o = tid), 8 b-rows per thread; x tile staged in LDS
// (uniform-per-lane LDS broadcast reads); grid = B/8 = 128 blocks.
// ---------------------------------------------------------------------------
__global__ void mmro_final(const float* __restrict__ x,
                           const float* __restrict__ w,
                           const float* __restrict__ relx,
                           const float* __restrict__ relw,
                           float* __restrict__ outx,
                           float* __restrict__ outw) {
    __shared__ float xs[8][64];
    const int o  = threadIdx.x;      // 0..255
    const int b0 = blockIdx.x * 8;

    float cw[8], bestx[8], bestw[8], vox[8], vow[8];
#pragma unroll
    for (int i = 0; i < 8; ++i) {
        cw[i]    = relw[(size_t)(b0 + i) * O_DIM + o];
        bestx[i] = 3.0e38f;
        bestw[i] = 3.0e38f;
        vox[i]   = 0.0f;
        vow[i]   = 0.0f;
    }

    for (int d0 = 0; d0 < D_DIM; d0 += 64) {
        // stage x[b0..b0+7][d0..d0+63] : 512 floats, 2 per thread, coalesced
        {
            const int k0 = threadIdx.x;
            const int k1 = threadIdx.x + 256;
            xs[k0 >> 6][k0 & 63] = x[(size_t)(b0 + (k0 >> 6)) * D_DIM + d0 + (k0 & 63)];
            xs[k1 >> 6][k1 & 63] = x[(size_t)(b0 + (k1 >> 6)) * D_DIM + d0 + (k1 & 63)];
        }
        if (d0 + 64 < D_DIM) {
            __builtin_prefetch(&w[(size_t)(d0 + 64) * O_DIM + o]);
            __builtin_prefetch(&relx[(size_t)(d0 + 64) * O_DIM + o]);
        }
        __syncthreads();
#pragma unroll 4
        for (int dd = 0; dd < 64; ++dd) {
            const int d = d0 + dd;
            const float wv  = w[(size_t)d * O_DIM + o];
            const float rxv = relx[(size_t)d * O_DIM + o];
#pragma unroll
            for (int i = 0; i < 8; ++i) {
                const float xv    = xs[i][dd];         // LDS broadcast
                const float inner = fmaxf(xv, wv);
                const float mx    = fmaxf(xv, rxv);
                if (mx < bestx[i]) { bestx[i] = mx; vox[i] = inner; }
                const float mw    = fmaxf(cw[i], wv);
                if (mw < bestw[i]) { bestw[i] = mw; vow[i] = inner; }
            }
        }
        __syncthreads();
    }
#pragma unroll
    for (int i = 0; i < 8; ++i) {
        outx[(size_t)(b0 + i) * O_DIM + o] = vox[i];
        outw[(size_t)(b0 + i) * O_DIM + o] = vow[i];
    }
}

extern "C" void kernel_launch(void* const* d_in, const int* in_sizes, int n_in,
                              void* d_out, int out_size, void* d_ws, size_t ws_size,
                              hipStream_t stream) {
    const float* x = (const float*)d_in[0];   // (B, D)
    const float* w = (const float*)d_in[1];   // (D, O)
    const float* t = (const float*)d_in[2];   // (B, O)
    float* out = (float*)d_out;               // chosen_x (B,O) then chosen_w (B,O)

    float* ws   = (float*)d_ws;
    float* Sx   = ws;                                  // 512
    float* Sw   = ws + 512;                            // 256
    float* relx = ws + 768;                            // D*O = 131072
    float* relw = ws + 768 + D_DIM * O_DIM;            // B*O = 262144

    // Phase 0: denominators via WMMA (32 waves for Sx, 16 for Sw)
    mmro_sums_wmma<<<48, 32, 0, stream>>>(x, w, Sx, Sw);
    // Phase 1: rel_x (D x O), needs Sx
    mmro_relx<<<128, 128, 0, stream>>>(x, t, Sx, relx);
    // Phase 2: rel_w (B x O), needs Sw
    mmro_relw<<<64, 256, 0, stream>>>(w, t, Sw, relw);
    // Phase 3: fused argmins + outputs
    mmro_final<<<128, 256, 0, stream>>>(x, w, relx, relw, out, out + B_DIM * O_DIM);
}